// Wasserstein_ADFWI_69320772157806
// MI455X (gfx1250) — compile-verified
//
#include <hip/hip_runtime.h>
#include <stdint.h>

#define T_LEN 2048
#define N_CH  256
#define S_DIM 8
#define B_DIM 2
#define EPS_F 1e-10f
#define DT_F  0.001f

typedef __attribute__((ext_vector_type(4))) unsigned int v4u;
typedef __attribute__((ext_vector_type(8))) int          v8i;
typedef __attribute__((ext_vector_type(4))) int          v4i;

__device__ __forceinline__ unsigned flip_f(float f) {
    unsigned u = __float_as_uint(f);
    return (u & 0x80000000u) ? ~u : (u ^ 0x80000000u);
}
__device__ __forceinline__ float unflip_f(unsigned u) {
    unsigned v = (u & 0x80000000u) ? (u ^ 0x80000000u) : ~u;
    return __uint_as_float(v);
}

// LDS address map produced by TDM padding (1 dword pad every 128 dwords):
// float index of element e is e + (e >> 7).
__device__ __forceinline__ int ladr(int e) { return e + (e >> 7); }

// ---------------------------------------------------------------------------
// TDM: one descriptor moves a [rows x tile_w] f32 tile (row stride in elems)
// from global to LDS, inserting 1 dword of padding every 128 dwords.
// ---------------------------------------------------------------------------
__device__ __forceinline__ void tdm_load_2d(uint32_t lds_byte, uint64_t gaddr,
                                            uint32_t tile_w, uint32_t rows,
                                            uint32_t row_stride_elems)
{
    v4u g0;
    g0.x = 1u;                                                   // count=1 (valid), user mode
    g0.y = lds_byte;                                             // LDS dest (bytes)
    g0.z = (uint32_t)gaddr;                                      // global_addr[31:0]
    g0.w = (uint32_t)((gaddr >> 32) & 0x1FFFFFFu) | (2u << 30);  // addr[56:32] | type=2
    v8i g1;
    // data_size=4B (2), pad_enable=1, pad_interval=6 (128 dwords), pad_amount=0 (1 dword)
    g1[0] = (int)((2u << 16) | (1u << 20) | (6u << 22));
    g1[1] = (int)(tile_w << 16);          // tensor_dim0 (bits 79:48, low half)
    g1[2] = (int)(rows << 16);            // tensor_dim0 hi = 0 | tensor_dim1 low16
    g1[3] = (int)(tile_w << 16);          // tensor_dim1 hi = 0 | tile_dim0
    g1[4] = (int)rows;                    // tile_dim1 | tile_dim2=0
    g1[5] = (int)row_stride_elems;        // tensor_dim0_stride[31:0]
    g1[6] = 0;                            // stride0 hi | tensor_dim1_stride lo
    g1[7] = 0;
    v4i z4 = {0, 0, 0, 0};
#if __has_include(<hip/amd_detail/amd_gfx1250_TDM.h>)
    v8i z8 = {0, 0, 0, 0, 0, 0, 0, 0};
    __builtin_amdgcn_tensor_load_to_lds(g0, g1, z4, z4, z8, 0);
#else
    __builtin_amdgcn_tensor_load_to_lds(g0, g1, z4, z4, 0);
#endif
}

// ---------------------------------------------------------------------------
// Kernel 2 (placed first so the disasm snippet shows the TDM sequence):
// per-trace W2 via merged-CDF quantile formula.
// grid = B*S*(N/4) = 1024 blocks, 256 threads.
// LDS: interleaved [t][trace] layout with TDM-inserted padding (bank-skewed),
//      2 arrays x (8192+64) dwords = 66048 bytes.
// Phase A: TENSOR_LOAD_TO_LDS (one DMA per array, wave 0 only, TENSORcnt).
// Phase B: wave32 shuffle scans (32-wide tiles + carry) -> CDFs in place.
// Phase C: chunked two-pointer merge walk, per-element W2 contributions.
// ---------------------------------------------------------------------------
__global__ __launch_bounds__(256) void wadfwi_w2_kernel(
    const float* __restrict__ x, const float* __restrict__ y,
    const float* __restrict__ Sx, const float* __restrict__ Sy,
    const unsigned* __restrict__ gmin, float* __restrict__ out)
{
    __shared__ float s_buf[2 * 4 * T_LEN + 128];  // 66048 bytes (incl. TDM padding)

    const int tid = threadIdx.x;
    const int g   = blockIdx.x;          // 0..1023
    const int n0  = (g & 63) * 4;
    const int bs  = g >> 6;              // 0..15
    const int b   = bs >> 3;             // / S_DIM
    const size_t gbase = (size_t)bs * T_LEN * N_CH;
    const int wid = tid >> 5, lane = tid & 31;

    // ---- Phase A: TDM global -> LDS (x tile, then y tile), wave 0 issues ----
    if (wid == 0) {
        const uint32_t lbase = (uint32_t)(uintptr_t)(&s_buf[0]);
        tdm_load_2d(lbase,
                    (uint64_t)(uintptr_t)(x + gbase + n0), 4u, (uint32_t)T_LEN, (uint32_t)N_CH);
        tdm_load_2d(lbase + (uint32_t)(ladr(4 * T_LEN) * sizeof(float)),   // byte 33024
                    (uint64_t)(uintptr_t)(y + gbase + n0), 4u, (uint32_t)T_LEN, (uint32_t)N_CH);
    }

    // Scalar fetches that do not depend on LDS: overlap with the DMA.
    const float mn    = unflip_f(*gmin);
    const float shift = -1.1f * mn;      // >= 0
    const int   trB   = wid & 3;
    const int   isY   = wid >> 2;        // 0: x/cu, 1: y/cv
    const int   mB    = bs * N_CH + n0 + trB;
    const float SrawB = isY ? Sy[mB] : Sx[mB];
    const int   trC   = tid >> 6;        // 0..3
    const int   mC    = bs * N_CH + n0 + trC;
    const float Sshx  = Sx[mC] + shift * (float)T_LEN;
    const float Sshy  = Sy[mC] + shift * (float)T_LEN;

    if (wid == 0) __builtin_amdgcn_s_wait_tensorcnt(0);
    __syncthreads();

    // ---- Phase B: per-wave scan -> normalized CDF in place ----
    {
        const int ebase = isY * (4 * T_LEN);
        const float inv = 1.0f / ((SrawB + shift * (float)T_LEN) + EPS_F);
        float carry = 0.f;
        for (int k = 0; k < T_LEN / 32; ++k) {
            const int t = k * 32 + lane;
            const int a = ladr(ebase + t * 4 + trB);
            const float v = s_buf[a];
            float sc = v;
            for (int d = 1; d < 32; d <<= 1) {
                float nb = __shfl_up(sc, d, 32);
                if (lane >= d) sc += nb;
            }
            s_buf[a] = ((carry + sc) + (float)(t + 1) * shift) * inv;
            carry += __shfl(sc, 31, 32);
        }
    }
    __syncthreads();

    // ---- Phase C: merged-quantile W2 contributions ----
    float acc = 0.f;
    {
        const int tr  = trC;
        const int sub = tid & 63;
        if (Sshx != 0.f && Sshy != 0.f) {
            auto A  = [&](int i) -> float { return s_buf[ladr((i << 2) + tr)]; };
            auto Bv = [&](int k) -> float { return s_buf[ladr(4 * T_LEN + (k << 2) + tr)]; };
            if (sub < 32) {
                // elements of cu; tie-break: cu before cv at equal values
                const int i0 = sub * 64;
                const float a0 = A(i0);
                int lo = 0, hi = T_LEN;
                while (lo < hi) { int mid = (lo + hi) >> 1; if (Bv(mid) < a0) lo = mid + 1; else hi = mid; }
                int r = lo;                              // #Bv < a
                float prev = (i0 > 0) ? A(i0 - 1) : 0.f;
                for (int i = i0; i < i0 + 64; ++i) {
                    const float a = A(i);
                    while (r < T_LEN && Bv(r) < a) ++r;
                    float pred = prev;
                    if (r > 0) pred = fmaxf(pred, Bv(r - 1));
                    int iu = i;
                    while (iu > 0 && A(iu - 1) == a) --iu;   // #A < a
                    const int ii = iu < T_LEN - 1 ? iu : T_LEN - 1;
                    const int rr = r  < T_LEN - 1 ? r  : T_LEN - 1;
                    const float d = (float)ii * DT_F - (float)rr * DT_F;
                    acc += (a - pred) * d * d;
                    prev = a;
                }
            } else {
                // elements of cv
                const int k0 = (sub - 32) * 64;
                const float b0 = Bv(k0);
                int lo = 0, hi = T_LEN;
                while (lo < hi) { int mid = (lo + hi) >> 1; if (A(mid) <= b0) lo = mid + 1; else hi = mid; }
                int s = lo;                              // #A <= b
                float prev = (k0 > 0) ? Bv(k0 - 1) : 0.f;
                for (int k = k0; k < k0 + 64; ++k) {
                    const float bb = Bv(k);
                    while (s < T_LEN && A(s) <= bb) ++s;
                    float pred = prev;
                    if (s > 0) pred = fmaxf(pred, A(s - 1));
                    int iu = s;
                    while (iu > 0 && A(iu - 1) == bb) --iu;  // #A < b
                    int iv = k;
                    while (iv > 0 && Bv(iv - 1) == bb) --iv; // #Bv < b
                    const int ii = iu < T_LEN - 1 ? iu : T_LEN - 1;
                    const int kk = iv < T_LEN - 1 ? iv : T_LEN - 1;
                    const float d = (float)ii * DT_F - (float)kk * DT_F;
                    acc += (bb - pred) * d * d;
                    prev = bb;
                }
            }
        }
    }
    __syncthreads();   // everyone done reading s_buf; safe to reuse for reduce

    for (int off = 16; off > 0; off >>= 1)
        acc += __shfl_down(acc, off, 32);
    if (lane == 0) s_buf[wid] = acc;
    __syncthreads();
    if (tid == 0) {
        float tot = 0.f;
        for (int i = 0; i < 8; ++i) tot += s_buf[i];
        atomicAdd(&out[b], tot);
    }
}

// ---------------------------------------------------------------------------
// Kernel 1: per-trace raw sums over T, and global min(x, y, 0).
// grid = (B*S, 16), block = 256 (thread == channel n). Fully coalesced.
// ---------------------------------------------------------------------------
__global__ __launch_bounds__(256) void wadfwi_reduce_kernel(
    const float* __restrict__ x, const float* __restrict__ y,
    float* __restrict__ Sx, float* __restrict__ Sy, unsigned* __restrict__ gmin)
{
    const int bs = blockIdx.x;   // 0..15  (b*S+s)
    const int tc = blockIdx.y;   // 0..15  (chunk of T)
    const int n  = threadIdx.x;  // 0..255
    const size_t base = (size_t)bs * T_LEN * N_CH;

    float sx = 0.f, sy = 0.f, mloc = 0.f;  // min includes 0 by construction
    const int t0 = tc * (T_LEN / 16);
    for (int t = t0; t < t0 + T_LEN / 16; ++t) {
        float vx = x[base + (size_t)t * N_CH + n];
        float vy = y[base + (size_t)t * N_CH + n];
        sx += vx; sy += vy;
        mloc = fminf(mloc, fminf(vx, vy));
    }
    atomicAdd(&Sx[bs * N_CH + n], sx);
    atomicAdd(&Sy[bs * N_CH + n], sy);

    for (int off = 16; off > 0; off >>= 1)
        mloc = fminf(mloc, __shfl_down(mloc, off, 32));
    __shared__ float smin[8];
    const int wid = threadIdx.x >> 5, lane = threadIdx.x & 31;
    if (lane == 0) smin[wid] = mloc;
    __syncthreads();
    if (threadIdx.x == 0) {
        float m = smin[0];
        for (int i = 1; i < 8; ++i) m = fminf(m, smin[i]);
        atomicMin(gmin, flip_f(m));
    }
}

// ---------------------------------------------------------------------------
extern "C" void kernel_launch(void* const* d_in, const int* in_sizes, int n_in,
                              void* d_out, int out_size, void* d_ws, size_t ws_size,
                              hipStream_t stream)
{
    (void)in_sizes; (void)n_in; (void)ws_size;
    const float* x = (const float*)d_in[0];
    const float* y = (const float*)d_in[1];
    float* out = (float*)d_out;

    unsigned* gmin = (unsigned*)d_ws;
    float* Sx = (float*)d_ws + 4;            // 16-byte offset into workspace
    float* Sy = Sx + (B_DIM * S_DIM * N_CH); // +4096 floats

    hipMemsetAsync(d_out, 0, (size_t)out_size * sizeof(float), stream);
    hipMemsetAsync((void*)Sx, 0, (size_t)2 * B_DIM * S_DIM * N_CH * sizeof(float), stream);
    hipMemsetAsync((void*)gmin, 0xFF, sizeof(unsigned), stream);

    dim3 g1(B_DIM * S_DIM, 16);
    wadfwi_reduce_kernel<<<g1, 256, 0, stream>>>(x, y, Sx, Sy, gmin);

    const int nblocks = B_DIM * S_DIM * (N_CH / 4);  // 1024
    wadfwi_w2_kernel<<<nblocks, 256, 0, stream>>>(x, y, Sx, Sy, gmin, out);
}